// IDWT_2963527434290
// MI455X (gfx1250) — compile-verified
//
#include <hip/hip_runtime.h>
#include <cstdint>

// Inverse Haar DWT, analytically folded to a per-pixel butterfly:
//   out[2h,2w]     = 0.5*(LL - LH - HL + HH)
//   out[2h,2w+1]   = 0.5*(LL + LH - HL - HH)
//   out[2h+1,2w]   = 0.5*(LL - LH + HL - HH)
//   out[2h+1,2w+1] = 0.5*(LL + LH + HL + HH)
// Memory-bound (512 MiB traffic, ~23 us floor @ 23.3 TB/s). Implemented as a
// persistent-block stream with double-buffered gfx1250 async global->LDS
// copies (ASYNCcnt) to overlap fetch of stage k+1 with compute of stage k.

namespace {
constexpr int kB = 16, kC = 64, kH = 128, kW = 128;
constexpr int kW4   = kW / 4;                       // 32 float4 per input row
constexpr int kNPos4 = kB * kC * kH * kW4;          // 4,194,304 float4 positions
constexpr int kBlock = 256;                         // 8 waves (wave32)
constexpr int kGrid  = 2048;                        // persistent blocks
constexpr int kIters = kNPos4 / (kBlock * kGrid);   // 8 stages per block (exact)
constexpr int kOutRow = 2 * kW;                     // 256 floats
constexpr int kOutImg = (2 * kH) * (2 * kW);        // 65536 floats per (b,c)
} // namespace

__global__ __launch_bounds__(kBlock) void idwt_haar_async(
    const float* __restrict__ ll_p, const float* __restrict__ lh_p,
    const float* __restrict__ hl_p, const float* __restrict__ hh_p,
    float* __restrict__ out)
{
    // 2 pipeline buffers x 4 subbands x 256 lanes x 16B = 32 KB LDS
    __shared__ float4 sbuf[2][4][kBlock];

    const unsigned t   = threadIdx.x;
    const unsigned bid = blockIdx.x;

    const float* srcs[4] = {ll_p, lh_p, hl_p, hh_p};

    // Issue 4 async global->LDS B128 copies (one per subband) for a stage.
    // Per-wave ASYNCcnt increases by 4. Lane<->LDS mapping is wave-private,
    // so only s_wait_asynccnt is needed before readback (no barriers).
    auto issue = [&](int buf, unsigned g4) {
#pragma unroll
        for (int i = 0; i < 4; ++i) {
            uint32_t la = (uint32_t)(uintptr_t)(&sbuf[buf][i][t]);
            const float* gp = srcs[i] + 4u * g4;
            asm volatile("global_load_async_to_lds_b128 %0, %1, off"
                         :: "v"(la), "v"(gp) : "memory");
        }
    };

    // Prologue: fetch stage 0 into buffer 0.
    issue(0, bid * kBlock + t);

    for (int k = 0; k < kIters; ++k) {
        const int cur = k & 1;
        const unsigned g4 = (bid + (unsigned)k * kGrid) * kBlock + t;

        if (k + 1 < kIters) {
            // Prefetch next stage, then wait until the current stage's
            // 4 async copies have landed in LDS (next stage's 4 remain).
            issue(cur ^ 1, (bid + (unsigned)(k + 1) * kGrid) * kBlock + t);
            asm volatile("s_wait_asynccnt 0x4" ::: "memory");
        } else {
            asm volatile("s_wait_asynccnt 0x0" ::: "memory");
        }

        // ds_load_b128 x4 from this lane's slots.
        float4 vll = sbuf[cur][0][t];
        float4 vlh = sbuf[cur][1][t];
        float4 vhl = sbuf[cur][2][t];
        float4 vhh = sbuf[cur][3][t];

        // Decode position: g4 = ((bc*128 + h)*32 + w4)
        const unsigned w4 = g4 & (kW4 - 1);
        const unsigned r  = g4 / kW4;
        const unsigned h  = r & (kH - 1);
        const unsigned bc = r / kH;

        float* o0 = out + (size_t)bc * kOutImg + (2u * h) * (unsigned)kOutRow + 8u * w4;
        float* o1 = o0 + kOutRow;

        const float* a4 = (const float*)&vll;
        const float* b4 = (const float*)&vlh;
        const float* c4 = (const float*)&vhl;
        const float* d4 = (const float*)&vhh;

        float r0[8], r1[8];
#pragma unroll
        for (int j = 0; j < 4; ++j) {
            const float a = a4[j], b = b4[j], c = c4[j], d = d4[j];
            const float s1 = a + d, s2 = b + c;   // a+d, b+c
            const float d1 = a - d, d2 = b - c;   // a-d, b-c
            r0[2 * j]     = 0.5f * (s1 - s2);     // LL-LH-HL+HH
            r0[2 * j + 1] = 0.5f * (d1 + d2);     // LL+LH-HL-HH
            r1[2 * j]     = 0.5f * (d1 - d2);     // LL-LH+HL-HH
            r1[2 * j + 1] = 0.5f * (s1 + s2);     // LL+LH+HL+HH
        }

        // Two b128 stores per output row; 32B-aligned, wave-coalesced.
        ((float4*)o0)[0] = float4{r0[0], r0[1], r0[2], r0[3]};
        ((float4*)o0)[1] = float4{r0[4], r0[5], r0[6], r0[7]};
        ((float4*)o1)[0] = float4{r1[0], r1[1], r1[2], r1[3]};
        ((float4*)o1)[1] = float4{r1[4], r1[5], r1[6], r1[7]};
    }
}

extern "C" void kernel_launch(void* const* d_in, const int* in_sizes, int n_in,
                              void* d_out, int out_size, void* d_ws, size_t ws_size,
                              hipStream_t stream) {
    (void)in_sizes; (void)n_in; (void)d_ws; (void)ws_size; (void)out_size;
    // setup_inputs order: LL, LH, HL, HH, matrix_low_0, matrix_low_1,
    // matrix_high_0, matrix_high_1. The Haar matrices are folded analytically.
    const float* LL = (const float*)d_in[0];
    const float* LH = (const float*)d_in[1];
    const float* HL = (const float*)d_in[2];
    const float* HH = (const float*)d_in[3];
    float* out = (float*)d_out;

    idwt_haar_async<<<kGrid, kBlock, 0, stream>>>(LL, LH, HL, HH, out);
}